// Pooler_10909216932176
// MI455X (gfx1250) — compile-verified
//
#include <hip/hip_runtime.h>
#include <math.h>

// Problem geometry (matches reference).
#define TOKENS   131072
#define HIDDEN   1024
#define SEGS     64

// Streaming geometry.
#define TPB      256                 // threads per block = 8 waves (wave32)
#define TT       8                   // tokens per LDS tile (8 * 4KB = 32KB)
#define CHUNK    128                 // tokens per block
#define NTILES   (CHUNK / TT)        // 16 tiles per block
#define NBLOCKS  (TOKENS / CHUNK)    // 1024 blocks
#define TILE_BYTES (TT * HIDDEN * 4) // 32768
#define PER_WAVE_ISSUES 8            // 32KB / (8 waves * 32 lanes * 16B)

// ---------------------------------------------------------------------------
// Kernel 0: zero the segment-sum accumulator in workspace (ws is poisoned).
// ---------------------------------------------------------------------------
__global__ void zero_ws_kernel(float* __restrict__ ws, int n) {
    int i = blockIdx.x * blockDim.x + threadIdx.x;
    if (i < n) ws[i] = 0.0f;
}

// ---------------------------------------------------------------------------
// Kernel 1: streamed segment sums.
// Double-buffered async global->LDS DMA (ASYNCcnt) + register accumulation,
// flushing to global f32 atomics only at (rare) segment boundaries.
// ---------------------------------------------------------------------------
__global__ __launch_bounds__(TPB) void pool_sum_kernel(
        const float* __restrict__ hs,
        const int*   __restrict__ seg,
        float*       __restrict__ sums) {
    __shared__ float buf[2][TT * HIDDEN];   // 2 * 32KB = 64KB static LDS

    const int tid  = threadIdx.x;
    const int wave = tid >> 5;
    const int lane = tid & 31;
    const int block0 = blockIdx.x * CHUNK;  // first token this block owns

    // 64-bit base of this block's token slab; all async offsets are 32-bit.
    const unsigned long long gbase =
        (unsigned long long)(const void*)(hs + (size_t)block0 * HIDDEN);

    // Raw LDS byte addresses (low 32 bits of the generic pointer = LDS offset).
    const unsigned ldsb0 = (unsigned)(size_t)(&buf[0][0]);
    const unsigned ldsb1 = (unsigned)(size_t)(&buf[1][0]);

    // Issue one 32KB tile as 8 async b128 wave-instructions per wave.
    auto issue_tile = [&](int tile, unsigned ldsbase) {
        const unsigned tb = (unsigned)tile * (unsigned)TILE_BYTES;
#pragma unroll
        for (int i = 0; i < PER_WAVE_ISSUES; ++i) {
            unsigned lin   = ((unsigned)(wave * PER_WAVE_ISSUES + i)) * 512u
                           + (unsigned)lane * 16u;
            unsigned laddr = ldsbase + lin;   // LDS destination byte address
            unsigned voff  = tb + lin;        // 32-bit global byte offset
            asm volatile("global_load_async_to_lds_b128 %0, %1, %2"
                         :: "v"(laddr), "v"(voff), "s"(gbase)
                         : "memory");
        }
    };

    // Prologue: fill both buffers.
    issue_tile(0, ldsb0);
    issue_tile(1, ldsb1);

    float4 acc = make_float4(0.0f, 0.0f, 0.0f, 0.0f);
    int cur = -1;
    const int f = tid * 4;                  // this thread's 4 features

    for (int t = 0; t < NTILES; ++t) {
        // Per-wave: async loads complete in order, so <=8 outstanding means
        // tile t's 8 transfers have landed; then sync all waves of the block.
        if (t + 1 < NTILES) {
            asm volatile("s_wait_asynccnt 0x8" ::: "memory");
        } else {
            asm volatile("s_wait_asynccnt 0x0" ::: "memory");
        }
        __syncthreads();

        const float* tilebuf = &buf[t & 1][0];
#pragma unroll
        for (int k = 0; k < TT; ++k) {
            const int tok = block0 + t * TT + k;
            const int s   = seg[tok];       // block-uniform, cache-resident
            const float4 v =
                *(const float4*)(tilebuf + k * HIDDEN + f);  // ds_load_b128
            if (s != cur) {                 // block-uniform branch (sorted ids)
                if (cur >= 0) {
                    float* dst = sums + (size_t)cur * HIDDEN + f;
                    atomicAdd(dst + 0, acc.x);
                    atomicAdd(dst + 1, acc.y);
                    atomicAdd(dst + 2, acc.z);
                    atomicAdd(dst + 3, acc.w);
                }
                acc = make_float4(0.0f, 0.0f, 0.0f, 0.0f);
                cur = s;
            }
            acc.x += v.x; acc.y += v.y; acc.z += v.z; acc.w += v.w;
        }
        __syncthreads();                    // everyone done reading buf[t&1]
        if (t + 2 < NTILES) issue_tile(t + 2, (t & 1) ? ldsb1 : ldsb0);
    }

    // Final flush of the trailing run.
    if (cur >= 0) {
        float* dst = sums + (size_t)cur * HIDDEN + f;
        atomicAdd(dst + 0, acc.x);
        atomicAdd(dst + 1, acc.y);
        atomicAdd(dst + 2, acc.z);
        atomicAdd(dst + 3, acc.w);
    }
}

// ---------------------------------------------------------------------------
// Kernel 2: mean + L2 normalize. Counts recovered by binary search on the
// sorted segment ids (no count atomics anywhere).
// ---------------------------------------------------------------------------
__device__ __forceinline__ int lower_bound_i32(const int* __restrict__ a,
                                               int n, int val) {
    int lo = 0, hi = n;
    while (lo < hi) {
        int mid = (lo + hi) >> 1;
        if (a[mid] < val) lo = mid + 1; else hi = mid;
    }
    return lo;
}

__global__ __launch_bounds__(256) void finalize_kernel(
        const float* __restrict__ sums,
        const int*   __restrict__ seg,
        float*       __restrict__ out) {
    __shared__ float red[256];
    __shared__ float denom_s;

    const int s   = blockIdx.x;
    const int tid = threadIdx.x;

    const int lo  = lower_bound_i32(seg, TOKENS, s);
    const int hi  = lower_bound_i32(seg, TOKENS, s + 1);
    const float cnt = fmaxf((float)(hi - lo), 1.0f);   // empty-segment guard

    const int f = tid * 4;
    const float4 v = *(const float4*)(sums + (size_t)s * HIDDEN + f);
    float4 mean;
    mean.x = v.x / cnt; mean.y = v.y / cnt;
    mean.z = v.z / cnt; mean.w = v.w / cnt;

    red[tid] = mean.x * mean.x + mean.y * mean.y
             + mean.z * mean.z + mean.w * mean.w;
    __syncthreads();
#pragma unroll
    for (int o = 128; o > 0; o >>= 1) {
        if (tid < o) red[tid] += red[tid + o];
        __syncthreads();
    }
    if (tid == 0) denom_s = fmaxf(sqrtf(red[0]), 1e-12f);
    __syncthreads();

    const float d = denom_s;
    float* dst = out + (size_t)s * HIDDEN + f;
    dst[0] = mean.x / d;
    dst[1] = mean.y / d;
    dst[2] = mean.z / d;
    dst[3] = mean.w / d;
}

// ---------------------------------------------------------------------------
// Host launcher.
// ---------------------------------------------------------------------------
extern "C" void kernel_launch(void* const* d_in, const int* in_sizes, int n_in,
                              void* d_out, int out_size, void* d_ws, size_t ws_size,
                              hipStream_t stream) {
    (void)in_sizes; (void)n_in; (void)out_size; (void)ws_size;

    const float* hs  = (const float*)d_in[0];   // [TOKENS, HIDDEN] f32
    const int*   seg = (const int*)d_in[1];     // [TOKENS] i32, sorted
    float*       out = (float*)d_out;           // [SEGS, HIDDEN] f32
    float*       sums = (float*)d_ws;           // [SEGS, HIDDEN] f32 scratch

    const int accN = SEGS * HIDDEN;
    zero_ws_kernel<<<(accN + 255) / 256, 256, 0, stream>>>(sums, accN);
    pool_sum_kernel<<<NBLOCKS, TPB, 0, stream>>>(hs, seg, sums);
    finalize_kernel<<<SEGS, 256, 0, stream>>>(sums, seg, out);
}